// DisplacedGTOExternalFieldBlock_6373731467891
// MI455X (gfx1250) — compile-verified
//
#include <hip/hip_runtime.h>
#include <hip/hip_bf16.h>

typedef float v2f __attribute__((ext_vector_type(2)));
typedef float v8f __attribute__((ext_vector_type(8)));

// out[n, p] = sum_f matrix[p, f] * field[batch[n], perm[f]], perm = [0,3,1,2], P = 32
// D(16x16) = A(16x4) x B(4x16) via V_WMMA_F32_16X16X4_F32; two WMMAs per 16-node
// tile cover p = 0..15 and p = 16..31. Store-bandwidth bound: 256 MB streaming out.
__global__ __launch_bounds__(256, 1) void gto_proj_wmma(
    const int* __restrict__ batch,
    const float* __restrict__ field,    // (N_GRAPHS, 4)
    const float* __restrict__ matrix,   // (32, 4) row-major
    float* __restrict__ out,            // (n_nodes, 32)
    int n_nodes, int n_full_tiles, int total_waves)
{
    const int lane = threadIdx.x & 31;
    // Wave index is uniform by construction; readfirstlane makes that explicit so
    // the tile loop runs entirely in SGPRs (s_cmp/s_cbranch, no exec churn).
    const int wave = __builtin_amdgcn_readfirstlane(
        blockIdx.x * (blockDim.x >> 5) + (threadIdx.x >> 5));

    const int h = lane >> 4;   // lane half: selects K-pair {0,1} vs {2,3}
    const int m = lane & 15;   // M (node-in-tile) for A; N (p-in-halftile) for B/D

    // B operand layout (4x16, K x N): lane L holds B[2h][m], B[2h+1][m]
    // with B[k][p] = matrix[p][k]. Loaded once per wave, reused every tile.
    const float* r0 = matrix + (m)      * 4 + 2 * h;   // p = 0..15
    const float* r1 = matrix + (16 + m) * 4 + 2 * h;   // p = 16..31
    v2f b0; b0.x = r0[0]; b0.y = r0[1];
    v2f b1; b1.x = r1[0]; b1.y = r1[1];

    const v8f c = {};  // zero accumulator

    // ---- Hot loop: full tiles only, zero per-iteration conditions ----
    for (int tile = wave; tile < n_full_tiles; tile += total_waves) {
        const int base = tile * 16;                       // SGPR

        const int g = batch[base + m];
        const float4 f = reinterpret_cast<const float4*>(field)[g];  // b128, L2-resident

        // nf_perm = field[:, [0,3,1,2]]:
        //   lanes 0-15 : A[M][0]=f.x, A[M][1]=f.w
        //   lanes 16-31: A[M][2]=f.y, A[M][3]=f.z
        v2f a;
        a.x = h ? f.y : f.x;   // v_cndmask, no EXEC change
        a.y = h ? f.z : f.w;

        v8f d0 = __builtin_amdgcn_wmma_f32_16x16x4_f32(
            /*neg_a=*/false, a, /*neg_b=*/false, b0,
            /*c_mod=*/(short)0, c, /*reuse_a=*/false, /*reuse_b=*/false);
        v8f d1 = __builtin_amdgcn_wmma_f32_16x16x4_f32(
            false, a, false, b1, (short)0, c, false, false);

        // D layout: VGPR r, lane L -> row = base + r + 8h, col = m (+16 for d1).
        // 16 coalesced NT b32 stores = exactly the 128 B/node payload (one s_clause).
        float* o = out + (size_t)(base + 8 * h) * 32 + m;
#pragma unroll
        for (int r = 0; r < 8; ++r) {
            __builtin_nontemporal_store(d0[r], o);
            __builtin_nontemporal_store(d1[r], o + 16);
            o += 32;
        }
    }

    // ---- Tail: at most one partial tile, handled by exactly one wave ----
    const int tail = n_nodes - n_full_tiles * 16;         // 0..15
    if (tail != 0 && wave == (n_full_tiles % total_waves)) {
        const int base = n_full_tiles * 16;

        int node = base + m;
        node = (node < n_nodes) ? node : (n_nodes - 1);   // clamp: EXEC all-1s at WMMA
        const int g = batch[node];
        const float4 f = reinterpret_cast<const float4*>(field)[g];

        v2f a;
        a.x = h ? f.y : f.x;
        a.y = h ? f.z : f.w;

        v8f d0 = __builtin_amdgcn_wmma_f32_16x16x4_f32(
            false, a, false, b0, (short)0, c, false, false);
        v8f d1 = __builtin_amdgcn_wmma_f32_16x16x4_f32(
            false, a, false, b1, (short)0, c, false, false);

        const int rowbase = base + 8 * h;
        float* o = out + (size_t)rowbase * 32 + m;
#pragma unroll
        for (int r = 0; r < 8; ++r) {
            if (rowbase + r < n_nodes) {
                __builtin_nontemporal_store(d0[r], o);
                __builtin_nontemporal_store(d1[r], o + 16);
            }
            o += 32;
        }
    }
}

extern "C" void kernel_launch(void* const* d_in, const int* in_sizes, int n_in,
                              void* d_out, int out_size, void* d_ws, size_t ws_size,
                              hipStream_t stream) {
    const int*   batch  = (const int*)  d_in[0];
    // d_in[1] = positions: dead code in the reference output, unused.
    const float* field  = (const float*)d_in[2];
    const float* matrix = (const float*)d_in[3];
    float*       out    = (float*)      d_out;

    const int n_nodes      = in_sizes[0];
    const int n_tiles      = (n_nodes + 15) / 16;
    const int n_full_tiles = n_nodes / 16;

    const int waves_per_block = 8;                       // 256 threads, wave32
    int blocks = (n_tiles + waves_per_block - 1) / waves_per_block;
    if (blocks > 2048) blocks = 2048;                    // grid-stride; ~16k waves saturates HBM
    if (blocks < 1)    blocks = 1;
    const int total_waves = blocks * waves_per_block;

    gto_proj_wmma<<<blocks, 256, 0, stream>>>(batch, field, matrix, out,
                                              n_nodes, n_full_tiles, total_waves);
}